// GCN_23862838297156
// MI455X (gfx1250) — compile-verified
//
#include <hip/hip_runtime.h>
#include <hip/hip_bf16.h>

#define HID 128
#define LDSW 132   // padded row stride (floats) -> conflict-free ds reads, keeps 16B alignment
#define EPS 1e-5f

typedef float v2f __attribute__((ext_vector_type(2)));
typedef float v8f __attribute__((ext_vector_type(8)));

__device__ inline void atomAddF(float* p, float v) { unsafeAtomicAdd(p, v); }

// -------- degree --------
__global__ void deg_kernel(const int* __restrict__ dst, float* __restrict__ deg, int nedges) {
    int e = blockIdx.x * blockDim.x + threadIdx.x;
    if (e < nedges) atomAddF(&deg[dst[e]], 1.0f);
}
__global__ void dis_kernel(float* __restrict__ deg, int n) {
    int i = blockIdx.x * blockDim.x + threadIdx.x;
    if (i < n) deg[i] = rsqrtf(deg[i] + 1.0f);
}

// -------- WMMA GEMM: C = act(A) @ W (+bias), optional fused per-column stats --------
// act = batchnorm(mean,rstd)+relu if bn_mean != null, else identity.
// Block: 256 threads = 8 waves; block computes 32 rows x 128 cols.
// Wave w owns cols [16w,16w+16) and TWO row tiles (0-15, 16-31) sharing each B fragment,
// so every pair of WMMAs reuses one W fetch (halves B traffic vs 1 tile/wave).
__global__ void __launch_bounds__(256)
gemm_wmma(const float* __restrict__ A, const float* __restrict__ W,
          const float* __restrict__ bias,
          const float* __restrict__ bn_mean, const float* __restrict__ bn_rstd,
          float* __restrict__ C,
          float* __restrict__ stat_sum, float* __restrict__ stat_sumsq,
          int nrows)
{
    __shared__ float As[32 * LDSW];
    const int row0 = blockIdx.x * 32;
    const int tid = threadIdx.x;

    // stage A tile (32x128) with optional bn+relu applied
    for (int i = tid; i < 32 * HID; i += 256) {
        int r = i >> 7, f = i & 127;
        float v = 0.0f;
        if (row0 + r < nrows) {
            v = A[(size_t)(row0 + r) * HID + f];
            if (bn_mean) v = fmaxf((v - bn_mean[f]) * bn_rstd[f], 0.0f);
        }
        As[r * LDSW + f] = v;
    }
    __syncthreads();

    const int wave = tid >> 5;
    const int lane = tid & 31;
    const int rlo  = lane & 15;
    const int khi  = (lane >> 4) << 1;   // 0 or 2
    const int col0 = wave * 16;

    v8f acc0 = {}, acc1 = {};
    #pragma unroll 4
    for (int k0 = 0; k0 < HID; k0 += 4) {
        v2f a0, a1, b;
        a0.x = As[rlo * LDSW + k0 + khi];
        a0.y = As[rlo * LDSW + k0 + khi + 1];
        a1.x = As[(rlo + 16) * LDSW + k0 + khi];
        a1.y = As[(rlo + 16) * LDSW + k0 + khi + 1];
        b.x = W[(size_t)(k0 + khi) * HID + col0 + rlo];
        b.y = W[(size_t)(k0 + khi + 1) * HID + col0 + rlo];
        acc0 = __builtin_amdgcn_wmma_f32_16x16x4_f32(
            false, a0, false, b, (short)0, acc0, false, false);
        acc1 = __builtin_amdgcn_wmma_f32_16x16x4_f32(
            false, a1, false, b, (short)0, acc1, false, false);
    }
    __syncthreads();   // all A reads done; reuse As to stage the output tile

    // D layout: vgpr j -> row mbase+j, col rlo
    const float bcol = bias ? bias[col0 + rlo] : 0.0f;
    const int mbase = (lane >> 4) * 8;
    float s = 0.0f, s2 = 0.0f;
    #pragma unroll
    for (int j = 0; j < 8; ++j) {
        float v0 = acc0[j] + bcol;
        float v1 = acc1[j] + bcol;
        As[(mbase + j) * LDSW + col0 + rlo]      = v0;
        As[(16 + mbase + j) * LDSW + col0 + rlo] = v1;
        if (row0 + mbase + j < nrows)      { s += v0; s2 += v0 * v0; }
        if (row0 + 16 + mbase + j < nrows) { s += v1; s2 += v1 * v1; }
    }
    if (stat_sum) {
        atomAddF(&stat_sum[col0 + rlo], s);
        atomAddF(&stat_sumsq[col0 + rlo], s2);
    }
    __syncthreads();

    // coalesced float4 stores: 32 rows x 32 float4 = 1024; 256 threads -> 4 each
    for (int i = tid; i < 32 * 32; i += 256) {
        int r = i >> 5, c = (i & 31) << 2;
        if (row0 + r < nrows) {
            float4 v = *(const float4*)&As[r * LDSW + c];
            *(float4*)&C[(size_t)(row0 + r) * HID + c] = v;
        }
    }
}

// -------- edge scatter: agg[dst] += h[src] * dis[src]*dis[dst]  (wave per edge) --------
__global__ void __launch_bounds__(256)
scatter_kernel(const float* __restrict__ h, const int* __restrict__ src,
               const int* __restrict__ dst, const float* __restrict__ dis,
               float* __restrict__ agg, int nedges)
{
    int t = blockIdx.x * blockDim.x + threadIdx.x;
    int e = t >> 5;
    if (e >= nedges) return;
    int l = t & 31;
    int s = src[e], d = dst[e];
    float norm = dis[s] * dis[d];
    const float4 hv = *(const float4*)(h + (size_t)s * HID + l * 4);
    float* out = agg + (size_t)d * HID + l * 4;
    atomAddF(out + 0, hv.x * norm);
    atomAddF(out + 1, hv.y * norm);
    atomAddF(out + 2, hv.z * norm);
    atomAddF(out + 3, hv.w * norm);
}

// -------- combine: pre += h*dis^2 + b (in place), accumulate per-feature stats --------
__global__ void __launch_bounds__(256)
combine_stats(const float* __restrict__ h, const float* __restrict__ dis,
              const float* __restrict__ bias, float* __restrict__ pre,
              float* __restrict__ stat_sum, float* __restrict__ stat_sumsq, int n)
{
    const int f = threadIdx.x & 127;
    const int half = threadIdx.x >> 7;
    const int base = blockIdx.x * 64;
    const float b = bias[f];
    float s = 0.0f, s2 = 0.0f;
    for (int i = half; i < 64; i += 2) {
        int r = base + i;
        if (r >= n) break;
        float dv = dis[r];
        size_t idx = (size_t)r * HID + f;
        float v = pre[idx] + h[idx] * dv * dv + b;
        pre[idx] = v;
        s += v; s2 += v * v;
    }
    __shared__ float red[256];
    red[threadIdx.x] = s;
    __syncthreads();
    if (half == 0) atomAddF(&stat_sum[f], red[f] + red[f + 128]);
    __syncthreads();
    red[threadIdx.x] = s2;
    __syncthreads();
    if (half == 0) atomAddF(&stat_sumsq[f], red[f] + red[f + 128]);
}

// -------- batchnorm finalize --------
__global__ void bn_finalize(const float* __restrict__ sum, const float* __restrict__ sumsq,
                            float* __restrict__ mean, float* __restrict__ rstd, float n)
{
    int f = threadIdx.x;
    float m = sum[f] / n;
    float v = sumsq[f] / n - m * m;
    mean[f] = m;
    rstd[f] = rsqrtf(v + EPS);
}

// -------- final projection: out = bn_relu(y) @ lin2_w + lin2_b  (wave per node) --------
__global__ void __launch_bounds__(256)
lin2_kernel(const float* __restrict__ y, const float* __restrict__ mean,
            const float* __restrict__ rstd, const float* __restrict__ w2,
            const float* __restrict__ b2, float* __restrict__ out, int n)
{
    int t = blockIdx.x * blockDim.x + threadIdx.x;
    int i = t >> 5;
    if (i >= n) return;
    int l = t & 31;
    float d0 = 0.0f, d1 = 0.0f;
    for (int f = l; f < HID; f += 32) {
        float v = fmaxf((y[(size_t)i * HID + f] - mean[f]) * rstd[f], 0.0f);
        d0 += v * w2[f * 2 + 0];
        d1 += v * w2[f * 2 + 1];
    }
    #pragma unroll
    for (int off = 16; off > 0; off >>= 1) {
        d0 += __shfl_down(d0, off, 32);
        d1 += __shfl_down(d1, off, 32);
    }
    if (l == 0) {
        out[i * 2 + 0] = d0 + b2[0];
        out[i * 2 + 1] = d1 + b2[1];
    }
}

extern "C" void kernel_launch(void* const* d_in, const int* in_sizes, int n_in,
                              void* d_out, int out_size, void* d_ws, size_t ws_size,
                              hipStream_t stream)
{
    const float* x      = (const float*)d_in[0];
    const int*   ei     = (const int*)  d_in[1];
    const float* W0     = (const float*)d_in[2];
    const float* b0     = (const float*)d_in[3];
    const float* Ws     = (const float*)d_in[4];
    const float* bs     = (const float*)d_in[5];
    const float* lin1_w = (const float*)d_in[6];
    const float* lin1_b = (const float*)d_in[7];
    const float* lin2_w = (const float*)d_in[8];
    const float* lin2_b = (const float*)d_in[9];

    const int N = in_sizes[0] / HID;
    const int E = in_sizes[1] / 2;
    const int* src = ei;
    const int* dst = ei + E;

    const size_t NH = (size_t)N * HID;
    float* ws    = (float*)d_ws;
    float* bufH  = ws;
    float* bufP0 = ws + NH;
    float* bufP1 = ws + 2 * NH;
    float* dis   = ws + 3 * NH;          // also used as deg accumulator
    float* stats = dis + N;              // 4 slots x {sum,sumsq,mean,rstd} x 128
    auto slot = [&](int i) { return stats + (size_t)i * 512; };

    const int gemmGrid  = (N + 31) / 32;
    const int edgeGrid  = (int)(((size_t)E * 32 + 255) / 256);
    const int combGrid  = (N + 63) / 64;

    // degrees -> deg_inv_sqrt
    hipMemsetAsync(dis, 0, (size_t)N * sizeof(float), stream);
    hipMemsetAsync(stats, 0, 4 * 512 * sizeof(float), stream);
    deg_kernel<<<(E + 255) / 256, 256, 0, stream>>>(dst, dis, E);
    dis_kernel<<<(N + 255) / 256, 256, 0, stream>>>(dis, N);

    // ---- GCN layer 0: input x (no bn) ----
    gemm_wmma<<<gemmGrid, 256, 0, stream>>>(x, W0, nullptr, nullptr, nullptr,
                                            bufH, nullptr, nullptr, N);
    hipMemsetAsync(bufP0, 0, NH * sizeof(float), stream);
    scatter_kernel<<<edgeGrid, 256, 0, stream>>>(bufH, src, dst, dis, bufP0, E);
    combine_stats<<<combGrid, 256, 0, stream>>>(bufH, dis, b0, bufP0,
                                                slot(0), slot(0) + 128, N);
    bn_finalize<<<1, 128, 0, stream>>>(slot(0), slot(0) + 128,
                                       slot(0) + 256, slot(0) + 384, (float)N);

    // ---- GCN layer 1 ----
    gemm_wmma<<<gemmGrid, 256, 0, stream>>>(bufP0, Ws, nullptr,
                                            slot(0) + 256, slot(0) + 384,
                                            bufH, nullptr, nullptr, N);
    hipMemsetAsync(bufP1, 0, NH * sizeof(float), stream);
    scatter_kernel<<<edgeGrid, 256, 0, stream>>>(bufH, src, dst, dis, bufP1, E);
    combine_stats<<<combGrid, 256, 0, stream>>>(bufH, dis, bs, bufP1,
                                                slot(1), slot(1) + 128, N);
    bn_finalize<<<1, 128, 0, stream>>>(slot(1), slot(1) + 128,
                                       slot(1) + 256, slot(1) + 384, (float)N);

    // ---- GCN layer 2 ----
    gemm_wmma<<<gemmGrid, 256, 0, stream>>>(bufP1, Ws + HID * HID, nullptr,
                                            slot(1) + 256, slot(1) + 384,
                                            bufH, nullptr, nullptr, N);
    hipMemsetAsync(bufP0, 0, NH * sizeof(float), stream);
    scatter_kernel<<<edgeGrid, 256, 0, stream>>>(bufH, src, dst, dis, bufP0, E);
    combine_stats<<<combGrid, 256, 0, stream>>>(bufH, dis, bs + HID, bufP0,
                                                slot(2), slot(2) + 128, N);
    bn_finalize<<<1, 128, 0, stream>>>(slot(2), slot(2) + 128,
                                       slot(2) + 256, slot(2) + 384, (float)N);

    // ---- lin1 (bn+relu on input, bias, fused stats) ----
    gemm_wmma<<<gemmGrid, 256, 0, stream>>>(bufP0, lin1_w, lin1_b,
                                            slot(2) + 256, slot(2) + 384,
                                            bufP1, slot(3), slot(3) + 128, N);
    bn_finalize<<<1, 128, 0, stream>>>(slot(3), slot(3) + 128,
                                       slot(3) + 256, slot(3) + 384, (float)N);

    // ---- lin2: bn+relu then project to 2 ----
    lin2_kernel<<<(int)(((size_t)N * 32 + 255) / 256), 256, 0, stream>>>(
        bufP1, slot(3) + 256, slot(3) + 384, lin2_w, lin2_b, (float*)d_out, N);
}